// SimpleAttn_40183714022071
// MI455X (gfx1250) — compile-verified
//
#include <hip/hip_runtime.h>

// Problem constants (B, T, C, H) = (2, 4096, 128, 2), fp32 everywhere.
#define BNUM 2
#define TSEQ 4096
#define CDIM 128
#define HNUM 2

typedef __attribute__((ext_vector_type(2))) float v2f;
typedef __attribute__((ext_vector_type(8))) float v8f;

// D = A(16x4 f32) x B(4x16 f32) + C(16x16 f32), full-precision CDNA5 WMMA.
__device__ __forceinline__ v8f wmma4(v2f a, v2f b, v8f c) {
  return __builtin_amdgcn_wmma_f32_16x16x4_f32(
      /*neg_a=*/false, a, /*neg_b=*/false, b,
      /*c_mod=*/(short)0, c, /*reuse_a=*/false, /*reuse_b=*/false);
}

// ---------------------------------------------------------------------------
// Stage 0: zero the G accumulator region of the workspace.
// ---------------------------------------------------------------------------
__global__ void SimpleAttn_zero_kernel(float* __restrict__ p, int n) {
  int i = blockIdx.x * blockDim.x + threadIdx.x;
  if (i < n) p[i] = 0.0f;
}

// ---------------------------------------------------------------------------
// Stage 1: G[b] = X_b^T @ X_b   (128x128 per batch), split-K x16 with
// float atomics to combine partials. 2 batches * 64 tiles * 16 slices = 2048
// waves -> 256 blocks of 8 waves.
// ---------------------------------------------------------------------------
__global__ void SimpleAttn_xtx_kernel(const float* __restrict__ X,
                                      float* __restrict__ G) {
  const int lane = threadIdx.x & 31;
  const int wid  = threadIdx.x >> 5;
  const int wg   = blockIdx.x * 8 + wid;      // 0..2047
  const int b    = wg >> 10;                  // / (64*16)
  const int rem  = wg & 1023;
  const int tile = rem >> 4;                  // 0..63
  const int slice = rem & 15;                 // 0..15
  const int m0 = (tile >> 3) << 4;
  const int n0 = (tile & 7) << 4;
  const int r = lane & 15;
  const int half = lane >> 4;

  const float* Xb = X + (size_t)b * TSEQ * CDIM;
  v8f acc = {};
  const int kbeg = slice * (TSEQ / 16);
  const int kend = kbeg + (TSEQ / 16);
  for (int k0 = kbeg; k0 < kend; k0 += 4) {
    const int kA = k0 + 2 * half;
    v2f a, bb;
    // A = X^T : A[m, k] = X[k, m]
    a.x  = Xb[(size_t)(kA + 0) * CDIM + m0 + r];
    a.y  = Xb[(size_t)(kA + 1) * CDIM + m0 + r];
    // B = X   : B[k, n] = X[k, n]
    bb.x = Xb[(size_t)(kA + 0) * CDIM + n0 + r];
    bb.y = Xb[(size_t)(kA + 1) * CDIM + n0 + r];
    acc = wmma4(a, bb, acc);
  }
  float* Gt = G + (size_t)b * CDIM * CDIM;
#pragma unroll
  for (int v = 0; v < 8; ++v) {
    const int row = v + 8 * half;             // C/D layout: rows v / v+8
    atomicAdd(&Gt[(size_t)(m0 + row) * CDIM + n0 + r], acc[v]);
  }
}

// ---------------------------------------------------------------------------
// Stage 2: per (b,h):
//   stage==0 : P[b,h] = W_qk_h^T @ G[b]        (128x128x128)
//   stage==1 : M[b,h] = P[b,h]  @ W_ov_h       (128x128x128)
// 4 (b,h) * 64 tiles = 256 waves -> 32 blocks of 8 waves.
// ---------------------------------------------------------------------------
__global__ void SimpleAttn_head_gemm_kernel(const float* __restrict__ Wqk,
                                            const float* __restrict__ Wov,
                                            const float* __restrict__ G,
                                            float* __restrict__ P,
                                            float* __restrict__ M,
                                            int stage) {
  const int lane = threadIdx.x & 31;
  const int wid  = threadIdx.x >> 5;
  const int wg   = blockIdx.x * 8 + wid;      // 0..255
  const int bh   = wg >> 6;                   // 0..3
  const int tile = wg & 63;
  const int b = bh >> 1;
  const int h = bh & 1;
  const int m0 = (tile >> 3) << 4;
  const int n0 = (tile & 7) << 4;
  const int r = lane & 15;
  const int half = lane >> 4;

  v8f acc = {};
  if (stage == 0) {
    // A[m,k] = Wqk[k, h*C + m]  (transposed, ld = C*H)
    const float* A  = Wqk + h * CDIM;
    const float* Bm = G + (size_t)b * CDIM * CDIM;
    for (int k0 = 0; k0 < CDIM; k0 += 4) {
      const int kA = k0 + 2 * half;
      v2f a, bb;
      a.x  = A[(size_t)(kA + 0) * (CDIM * HNUM) + m0 + r];
      a.y  = A[(size_t)(kA + 1) * (CDIM * HNUM) + m0 + r];
      bb.x = Bm[(size_t)(kA + 0) * CDIM + n0 + r];
      bb.y = Bm[(size_t)(kA + 1) * CDIM + n0 + r];
      acc = wmma4(a, bb, acc);
    }
    float* D = P + (size_t)bh * CDIM * CDIM;
#pragma unroll
    for (int v = 0; v < 8; ++v)
      D[(size_t)(m0 + v + 8 * half) * CDIM + n0 + r] = acc[v];
  } else {
    // A[m,k] = P[bh][m, k] (row-major -> contiguous 8B per-lane loads)
    const float* A  = P + (size_t)bh * CDIM * CDIM;
    const float* Bm = Wov + h * CDIM;          // B[k,n] = Wov[k, h*C + n]
    for (int k0 = 0; k0 < CDIM; k0 += 4) {
      const int kA = k0 + 2 * half;
      v2f a = *(const v2f*)&A[(size_t)(m0 + r) * CDIM + kA];
      v2f bb;
      bb.x = Bm[(size_t)(kA + 0) * (CDIM * HNUM) + n0 + r];
      bb.y = Bm[(size_t)(kA + 1) * (CDIM * HNUM) + n0 + r];
      acc = wmma4(a, bb, acc);
    }
    float* D = M + (size_t)bh * CDIM * CDIM;
#pragma unroll
    for (int v = 0; v < 8; ++v)
      D[(size_t)(m0 + v + 8 * half) * CDIM + n0 + r] = acc[v];
  }
}

// ---------------------------------------------------------------------------
// Stage 3 (dominant, bandwidth-bound): out[b, t, h, :] = X_b[t, :] @ M[b,h].
// Each wave computes a 64x16 output slab (4 vertically stacked 16x16 tiles)
// so one B fragment feeds 4 WMMAs with independent accumulator chains.
// 4 (b,h) * 64 t-supertiles * 8 c-tiles = 2048 waves -> 256 blocks x 8 waves.
// Streams X once (4 MB), writes out once (8 MB); M (64 KB/head) lives in L2.
// ---------------------------------------------------------------------------
__global__ void SimpleAttn_out_gemm_kernel(const float* __restrict__ X,
                                           const float* __restrict__ M,
                                           float* __restrict__ Out) {
  const int lane = threadIdx.x & 31;
  const int wid  = threadIdx.x >> 5;
  const int wg   = blockIdx.x * 8 + wid;      // 0..2047
  const int bh   = wg >> 9;                   // 0..3
  const int rem  = wg & 511;
  const int ts   = rem >> 3;                  // 0..63 (t supertile, 64 rows)
  const int tn   = rem & 7;                   // 0..7  (c tile)
  const int b = bh >> 1;
  const int h = bh & 1;
  const int t0 = ts << 6;                     // 64 rows per supertile
  const int c0 = tn << 4;
  const int r = lane & 15;
  const int half = lane >> 4;

  const float* Xb = X + (size_t)b * TSEQ * CDIM;
  const float* Mm = M + (size_t)bh * CDIM * CDIM;
  const float* Arow0 = Xb + (size_t)(t0 +  0 + r) * CDIM;
  const float* Arow1 = Xb + (size_t)(t0 + 16 + r) * CDIM;
  const float* Arow2 = Xb + (size_t)(t0 + 32 + r) * CDIM;
  const float* Arow3 = Xb + (size_t)(t0 + 48 + r) * CDIM;

  v8f acc0 = {}, acc1 = {}, acc2 = {}, acc3 = {};
#pragma unroll 4
  for (int k0 = 0; k0 < CDIM; k0 += 4) {
    const int kA = k0 + 2 * half;
    // One B fragment shared by 4 WMMAs
    v2f bb;
    bb.x = Mm[(size_t)(kA + 0) * CDIM + c0 + r];
    bb.y = Mm[(size_t)(kA + 1) * CDIM + c0 + r];
    // Four A fragments: contiguous 8-byte loads per lane
    v2f a0 = *(const v2f*)&Arow0[kA];
    v2f a1 = *(const v2f*)&Arow1[kA];
    v2f a2 = *(const v2f*)&Arow2[kA];
    v2f a3 = *(const v2f*)&Arow3[kA];
    acc0 = wmma4(a0, bb, acc0);
    acc1 = wmma4(a1, bb, acc1);
    acc2 = wmma4(a2, bb, acc2);
    acc3 = wmma4(a3, bb, acc3);
  }
  // Out[b, t, h, c], t-row stride = H*C
  float* Dbase = Out + (((size_t)b * TSEQ + t0) * HNUM + h) * CDIM + c0 + r;
#pragma unroll
  for (int v = 0; v < 8; ++v) {
    const size_t row = (size_t)(v + 8 * half) * (HNUM * CDIM);
    Dbase[row                          ] = acc0[v];
    Dbase[row + 16 * (HNUM * CDIM)     ] = acc1[v];
    Dbase[row + 32 * (HNUM * CDIM)     ] = acc2[v];
    Dbase[row + 48 * (HNUM * CDIM)     ] = acc3[v];
  }
}

// ---------------------------------------------------------------------------
// Launch wiring. d_in[0]=input_seq (B,T,C) f32, d_in[1]=w_qk (C,C*H) f32,
// d_in[2]=w_ov (C,C*H) f32. d_out = (B,T,H,C) f32.
// Workspace layout (floats): G[2*128*128] | P[4*128*128] | M[4*128*128]
//   = 32768 + 65536 + 65536 = 163840 floats = 640 KB.
// ---------------------------------------------------------------------------
extern "C" void kernel_launch(void* const* d_in, const int* in_sizes, int n_in,
                              void* d_out, int out_size, void* d_ws, size_t ws_size,
                              hipStream_t stream) {
  (void)in_sizes; (void)n_in; (void)out_size; (void)ws_size;
  const float* X   = (const float*)d_in[0];
  const float* Wqk = (const float*)d_in[1];
  const float* Wov = (const float*)d_in[2];
  float* Out = (float*)d_out;

  float* G = (float*)d_ws;                       // 2 * 128 * 128
  float* P = G + BNUM * CDIM * CDIM;             // 4 * 128 * 128
  float* M = P + BNUM * HNUM * CDIM * CDIM;      // 4 * 128 * 128

  const int gN = BNUM * CDIM * CDIM;             // 32768
  SimpleAttn_zero_kernel<<<(gN + 255) / 256, 256, 0, stream>>>(G, gN);
  SimpleAttn_xtx_kernel<<<256, 256, 0, stream>>>(X, G);
  SimpleAttn_head_gemm_kernel<<<32, 256, 0, stream>>>(Wqk, Wov, G, P, M, 0);
  SimpleAttn_head_gemm_kernel<<<32, 256, 0, stream>>>(Wqk, Wov, G, P, M, 1);
  SimpleAttn_out_gemm_kernel<<<256, 256, 0, stream>>>(X, M, Out);
}